// HashEmbedding_71012989272119
// MI455X (gfx1250) — compile-verified
//
#include <hip/hip_runtime.h>
#include <cmath>

#define N_LEVELS   16
#define LOG2_T     19
#define TABLE_SIZE (1u << LOG2_T)
#define TABLE_MASK (TABLE_SIZE - 1u)
#define N_POINTS   (1u << 20)
#define PRIME1     2654435761u
#define PRIME2     805459861u

typedef __attribute__((ext_vector_type(2))) float v2f;

struct ResArr { float r[N_LEVELS]; };

// One thread per (point, level). tid = point*16 + level, so the float2 output
// index is exactly tid -> fully coalesced 256B stores per wave32.
__global__ __launch_bounds__(256) void hashgrid_trilerp_kernel(
    const float* __restrict__ x,
    const float* __restrict__ tables,
    float* __restrict__ out,
    ResArr res)
{
    const unsigned tid   = blockIdx.x * blockDim.x + threadIdx.x;
    const unsigned point = tid >> 4;
    const unsigned level = tid & 15u;

    // Point coords: 16 lanes share one point; L0 broadcasts the 12B.
    const float* xp = x + (size_t)point * 3u;
    const float px = xp[0];
    const float py = xp[1];
    const float pz = xp[2];

    // xn = (x - X_MIN) / (X_MAX - X_MIN), X_MIN=-1, X_MAX=1
    const float nx = (px + 1.0f) * 0.5f;
    const float ny = (py + 1.0f) * 0.5f;
    const float nz = (pz + 1.0f) * 0.5f;

    const float r  = res.r[level];
    const float sx = nx * r;
    const float sy = ny * r;
    const float sz = nz * r;

    const float fx = floorf(sx);
    const float fy = floorf(sy);
    const float fz = floorf(sz);

    const float wx = sx - fx;   // xd
    const float wy = sy - fy;   // yd
    const float wz = sz - fz;   // zd

    const int vx = (int)fx;
    const int vy = (int)fy;
    const int vz = (int)fz;

    // Per-axis prime products: 8 corner hashes become pure XORs.
    const unsigned hx0 = (unsigned)vx;                     // * PRIMES[0] == 1
    const unsigned hx1 = (unsigned)(vx + 1);
    const unsigned hy0 = (unsigned)vy        * PRIME1;
    const unsigned hy1 = (unsigned)(vy + 1)  * PRIME1;
    const unsigned hz0 = (unsigned)vz        * PRIME2;
    const unsigned hz1 = (unsigned)(vz + 1)  * PRIME2;

    const unsigned h00 = hy0 ^ hz0;
    const unsigned h01 = hy0 ^ hz1;
    const unsigned h10 = hy1 ^ hz0;
    const unsigned h11 = hy1 ^ hz1;

    // OFFSETS order: n = (i<<2)|(j<<1)|k with i->x, j->y, k->z
    const unsigned i0 = (hx0 ^ h00) & TABLE_MASK;  // (0,0,0)
    const unsigned i1 = (hx0 ^ h01) & TABLE_MASK;  // (0,0,1)
    const unsigned i2 = (hx0 ^ h10) & TABLE_MASK;  // (0,1,0)
    const unsigned i3 = (hx0 ^ h11) & TABLE_MASK;  // (0,1,1)
    const unsigned i4 = (hx1 ^ h00) & TABLE_MASK;  // (1,0,0)
    const unsigned i5 = (hx1 ^ h01) & TABLE_MASK;  // (1,0,1)
    const unsigned i6 = (hx1 ^ h10) & TABLE_MASK;  // (1,1,0)
    const unsigned i7 = (hx1 ^ h11) & TABLE_MASK;  // (1,1,1)

    // 8 independent global_load_b64 gathers -> deep memory-level parallelism.
    // Tables (64MB) live in the 192MB L2; these stay default-temporal.
    const v2f* __restrict__ tab = (const v2f*)tables + (size_t)level * TABLE_SIZE;
    const v2f e0 = tab[i0];
    const v2f e1 = tab[i1];
    const v2f e2 = tab[i2];
    const v2f e3 = tab[i3];
    const v2f e4 = tab[i4];
    const v2f e5 = tab[i5];
    const v2f e6 = tab[i6];
    const v2f e7 = tab[i7];

    // Exact lerp chain of the reference (c00/c01/c10/c11 -> c0/c1 -> c).
    const float omx = 1.0f - wx;
    const float omy = 1.0f - wy;
    const float omz = 1.0f - wz;

    const v2f c00 = e0 * omx + e4 * wx;
    const v2f c01 = e1 * omx + e5 * wx;
    const v2f c10 = e2 * omx + e6 * wx;
    const v2f c11 = e3 * omx + e7 * wx;

    const v2f c0 = c00 * omy + c10 * wy;
    const v2f c1 = c01 * omy + c11 * wy;

    const v2f c = c0 * omz + c1 * wz;

    // Non-temporal (TH=NT) store: the 128MB output stream must not evict the
    // L2-resident tables. Index == tid -> coalesced 256B per wave.
    __builtin_nontemporal_store(c, (v2f*)out + tid);
}

extern "C" void kernel_launch(void* const* d_in, const int* in_sizes, int n_in,
                              void* d_out, int out_size, void* d_ws, size_t ws_size,
                              hipStream_t stream) {
    const float* x      = (const float*)d_in[0];
    const float* tables = (const float*)d_in[1];
    float*       out    = (float*)d_out;

    // Bit-exact replication of numpy's float64 resolution computation:
    // _B = exp((log(512) - log(16)) / 15); res[l] = floor(16 * _B**l)
    ResArr res;
    const double b = exp((log(512.0) - log(16.0)) / 15.0);
    for (int l = 0; l < N_LEVELS; ++l) {
        res.r[l] = (float)floor(16.0 * pow(b, (double)l));
    }

    const unsigned total_threads = N_POINTS * N_LEVELS;   // 16,777,216
    const dim3 block(256);
    const dim3 grid(total_threads / 256);                 // 65,536 blocks

    hipLaunchKernelGGL(hashgrid_trilerp_kernel, grid, block, 0, stream,
                       x, tables, out, res);
}